// CapsulePooling2D_33638183862997
// MI455X (gfx1250) — compile-verified
//
#include <hip/hip_runtime.h>
#include <hip/hip_bf16.h>
#include <math.h>

typedef float v2f __attribute__((ext_vector_type(2)));
typedef float v4f __attribute__((ext_vector_type(4)));
typedef float v8f __attribute__((ext_vector_type(8)));

#define CP_EPS 1e-7f

// x: (32,64,64,256) f32 NHWC  ->  out: (32,32,32,256) f32
// Each wave handles 4 pooling blocks (2x2 spatial x 256 ch).
// Gram G16 = X * X^T via 64x V_WMMA_F32_16X16X4_F32 (A operand == B operand),
// two independent accumulator chains to relax the WMMA RAW chain,
// then tiny per-block routing math, then weighted-average output pass.
__global__ __launch_bounds__(256) void capsule_pool2d_kernel(
    const float* __restrict__ x, float* __restrict__ out) {
  const int lane = threadIdx.x & 31;
  const int waveInWg = threadIdx.x >> 5;
  const int wid = blockIdx.x * 8 + waveInWg;  // 8 waves / WG
  const int blk0 = wid * 4;                   // first pooling block of this wave

  __shared__ float Gbuf[8][16][17];  // per-wave 16x16 D scratch (+pad)
  __shared__ float w3q[8][4][4];     // per-wave, per-block 0.25*w3 weights

  // ---- Pass 1: accumulate G16 = X * X^T over K=256 in chunks of 4 ----
  const int r  = lane & 15;   // A-matrix row (pixel index among 16)
  const int hi = lane >> 4;   // 0 -> K{0,1}, 1 -> K{2,3} of each chunk
  const int beta_r = r >> 2;  // which of my wave's 4 blocks
  const int pix    = r & 3;   // pixel within 2x2 block
  {
    const int block = blk0 + beta_r;         // 0..32767 = b*1024 + i*32 + j
    const int b  = block >> 10;
    const int ij = block & 1023;
    const int i  = ij >> 5;
    const int j  = ij & 31;
    const int h  = 2 * i + (pix >> 1);
    const int w  = 2 * j + (pix & 1);
    const float* rowp =
        x + (((size_t)b * 64 + h) * 64 + w) * 256 + hi * 2;

    v8f acc0 = {};
    v8f acc1 = {};
    // 16 chunks of K=4 per outer iteration = two 128B cachelines per row;
    // chunks 0..7 feed acc0, chunks 8..15 feed acc1 (independent chains).
    for (int k = 0; k < 64; k += 16) {
#pragma unroll
      for (int u = 0; u < 8; ++u) {
        v2f a = *(const v2f*)(rowp + 4 * (k + u));       // global_load_b64
        acc0 = __builtin_amdgcn_wmma_f32_16x16x4_f32(
            false, a, false, a, (short)0, acc0, false, false);
        v2f c = *(const v2f*)(rowp + 4 * (k + 8 + u));   // global_load_b64
        acc1 = __builtin_amdgcn_wmma_f32_16x16x4_f32(
            false, c, false, c, (short)0, acc1, false, false);
      }
    }
    v8f acc = acc0 + acc1;

    // D layout: VGPR t, lanes 0-15 -> M=t, N=lane ; lanes 16-31 -> M=t+8
#pragma unroll
    for (int t = 0; t < 8; ++t) Gbuf[waveInWg][t + hi * 8][r] = acc[t];
  }
  __syncthreads();

  // ---- Pass 2: per-block routing scalar math (4 lanes, 4 blocks) ----
  if (lane < 4) {
    const int beta = lane;
    float g[4][4];
#pragma unroll
    for (int p = 0; p < 4; ++p)
#pragma unroll
      for (int q = 0; q < 4; ++q)
        g[p][q] = Gbuf[waveInWg][4 * beta + p][4 * beta + q];

    float s[4];
    // step 1: weights = 1
    {
      float Gw[4], n = 0.f;
#pragma unroll
      for (int p = 0; p < 4; ++p) {
        Gw[p] = g[p][0] + g[p][1] + g[p][2] + g[p][3];
        n += Gw[p];
      }
      n *= 0.0625f;                          // ||avg_fm||^2
      const float inv = 0.25f / (1.f + sqrtf(n) + CP_EPS);
#pragma unroll
      for (int p = 0; p < 4; ++p) s[p] = Gw[p] * inv;
    }
    // step 2: weights = normalize_pool_map(s1)
    {
      const float m = fmaxf(fmaxf(s[0], s[1]), fmaxf(s[2], s[3]));
      float e[4], se = 0.f;
#pragma unroll
      for (int p = 0; p < 4; ++p) { e[p] = expf((s[p] - m) * 0.5f); se += e[p]; }
      const float winv = 1.f / (se * 0.25f + CP_EPS);
      float wv[4];
#pragma unroll
      for (int p = 0; p < 4; ++p) wv[p] = e[p] * winv;
      float Gw[4], n = 0.f;
#pragma unroll
      for (int p = 0; p < 4; ++p) {
        Gw[p] = g[p][0]*wv[0] + g[p][1]*wv[1] + g[p][2]*wv[2] + g[p][3]*wv[3];
        n += Gw[p] * wv[p];
      }
      n *= 0.0625f;
      const float inv = 0.25f / (1.f + sqrtf(n) + CP_EPS);
#pragma unroll
      for (int p = 0; p < 4; ++p) s[p] += Gw[p] * inv;
    }
    // step 3 weights (fold final avg-pool's 0.25 in)
    {
      const float m = fmaxf(fmaxf(s[0], s[1]), fmaxf(s[2], s[3]));
      float e[4], se = 0.f;
#pragma unroll
      for (int p = 0; p < 4; ++p) { e[p] = expf((s[p] - m) * 0.5f); se += e[p]; }
      const float winv = 0.25f / (se * 0.25f + CP_EPS);
#pragma unroll
      for (int p = 0; p < 4; ++p) w3q[waveInWg][beta][p] = e[p] * winv;
    }
  }
  __syncthreads();

  // ---- Pass 3: out[c] = sum_p (0.25*w3[p]) * x[p][c]  (x re-read hits L2) ----
#pragma unroll
  for (int beta = 0; beta < 4; ++beta) {
    const int block = blk0 + beta;
    const int b  = block >> 10;
    const int ij = block & 1023;
    const int i  = ij >> 5;
    const int j  = ij & 31;
    const float w0 = w3q[waveInWg][beta][0];
    const float w1 = w3q[waveInWg][beta][1];
    const float w2 = w3q[waveInWg][beta][2];
    const float w3 = w3q[waveInWg][beta][3];
    const float* p00 = x + (((size_t)b * 64 + 2 * i) * 64 + 2 * j) * 256;
    const float* p01 = p00 + 256;
    const float* p10 = p00 + 64 * 256;
    const float* p11 = p10 + 256;
    float* op = out + (((size_t)b * 32 + i) * 32 + j) * 256;
#pragma unroll
    for (int t = 0; t < 2; ++t) {
      const int c = 4 * lane + 128 * t;     // coalesced float4 lanes
      v4f a0 = *(const v4f*)(p00 + c);
      v4f a1 = *(const v4f*)(p01 + c);
      v4f a2 = *(const v4f*)(p10 + c);
      v4f a3 = *(const v4f*)(p11 + c);
      v4f o = a0 * w0 + a1 * w1 + a2 * w2 + a3 * w3;
      *(v4f*)(op + c) = o;
    }
  }
}

extern "C" void kernel_launch(void* const* d_in, const int* in_sizes, int n_in,
                              void* d_out, int out_size, void* d_ws, size_t ws_size,
                              hipStream_t stream) {
  const float* x = (const float*)d_in[0];
  float* out = (float*)d_out;
  // 32768 pooling blocks / (4 per wave * 8 waves per WG) = 1024 workgroups
  capsule_pool2d_kernel<<<dim3(1024), dim3(256), 0, stream>>>(x, out);
}